// GAT_47459388621602
// MI455X (gfx1250) — compile-verified
//
#include <hip/hip_runtime.h>
#include <hip/hip_bf16.h>
#include <stdint.h>

// ---------------- constants (match reference) ----------------
#define N_NODES 50000
#define N_EDGES 800000
#define E_TOT   (N_EDGES + N_NODES)   // with self loops
#define IN_F    129
#define IN_PAD  132                   // pad K to multiple of 4 for WMMA f32 16x16x4
#define HID     32
#define HEADS   4
#define N_CLS   40
#define H1DIM   (HEADS * HID)         // 128
#define H2DIM   (HEADS * N_CLS)       // 160
#define NEG_SLOPE 0.2f
#define BN_EPS  1e-5f

typedef float v8f __attribute__((ext_vector_type(8)));
typedef float v2f __attribute__((ext_vector_type(2)));

// ---------------- helpers ----------------
__device__ inline void atomicMaxFloat(float* addr, float val) {
    // addr must be initialized to -inf; mixed signed-max / unsigned-min trick
    if (val >= 0.0f) {
        atomicMax((int*)addr, __float_as_int(val));
    } else {
        atomicMin((unsigned int*)addr, __float_as_uint(val));
    }
}

// ---------------- BatchNorm (eval) + pad to 132 cols ----------------
__global__ void bn_pad_kernel(const float* __restrict__ x,
                              const float* __restrict__ g, const float* __restrict__ be,
                              const float* __restrict__ mu, const float* __restrict__ var,
                              float* __restrict__ out) {
    int idx = blockIdx.x * blockDim.x + threadIdx.x;
    if (idx >= N_NODES * IN_PAD) return;
    int n = idx / IN_PAD;
    int f = idx - n * IN_PAD;
    float v = 0.0f;
    if (f < IN_F) {
        float xv = x[(size_t)n * IN_F + f];
        v = (xv - mu[f]) * rsqrtf(var[f] + BN_EPS) * g[f] + be[f];
    }
    out[idx] = v;
}

// ---------------- zero-pad W1 [129,128] -> [132,128] ----------------
__global__ void pad_w_kernel(const float* __restrict__ w, float* __restrict__ wp) {
    int idx = blockIdx.x * blockDim.x + threadIdx.x;
    if (idx >= IN_PAD * H1DIM) return;
    int r = idx / H1DIM;
    wp[idx] = (r < IN_F) ? w[idx] : 0.0f;
}

// ---------------- f32 WMMA GEMM: C[M,NCOLS] = A[M,K] @ W[K,NCOLS] ----------------
// One 16-row M tile per 128-thread block (4 waves). A strip (contiguous 16*K floats)
// staged into LDS via CDNA5 async-to-LDS loads; W pre-padded so inner loop is guard-free.
template<int K, int NCOLS>
__global__ void __launch_bounds__(128) gemm_wmma(const float* __restrict__ A,
                                                 const float* __restrict__ W,
                                                 float* __restrict__ C) {
    __shared__ float As[16 * K];
    const int m0 = blockIdx.x * 16;
    const int tid = threadIdx.x;

    // Async contiguous copy: global A[m0*K .. m0*K + 16K) -> LDS As[0 .. 16K)
    // (16 bytes per lane per issue; low 32 bits of a flat LDS pointer are the
    //  wave-relative LDS offset, which is what VDST of async-to-LDS expects)
    for (int i = tid * 4; i < 16 * K; i += 128 * 4) {
        uint32_t lds = (uint32_t)(uintptr_t)(&As[i]);
        const float* gp = A + (size_t)m0 * K + i;
        asm volatile("global_load_async_to_lds_b128 %0, %1, off"
                     :: "v"(lds), "v"(gp) : "memory");
    }
    asm volatile("s_wait_asynccnt 0x0" ::: "memory");
    __syncthreads();

    const int lane = tid & 31;
    const int wave = tid >> 5;
    const int mrow = lane & 15;          // A: lanes 0-15 and 16-31 both map M=0..15
    const int koff = (lane >> 4) << 1;   // A/B: VGPR pair covers K = k0+koff, k0+koff+1
    const int NT = NCOLS / 16;

    for (int nt = wave; nt < NT; nt += 4) {
        const int col = nt * 16 + (lane & 15);
        v8f c = {};
#pragma unroll
        for (int k0 = 0; k0 < K; k0 += 4) {
            v2f a;
            a.x = As[mrow * K + k0 + koff];
            a.y = As[mrow * K + k0 + koff + 1];
            const int r0 = k0 + koff;
            v2f b;
            b.x = W[(size_t)r0       * NCOLS + col];
            b.y = W[(size_t)(r0 + 1) * NCOLS + col];
            c = __builtin_amdgcn_wmma_f32_16x16x4_f32(false, a, false, b,
                                                      (short)0, c, false, false);
        }
        const int orow = m0 + ((lane >> 4) << 3);   // lanes 16-31 hold M = j+8
#pragma unroll
        for (int j = 0; j < 8; ++j) {
            C[(size_t)(orow + j) * NCOLS + col] = c[j];
        }
    }
}

// ---------------- per-node attention projections + softmax-state init ----------------
__global__ void alpha_kernel(const float* __restrict__ h,
                             const float* __restrict__ a_src, const float* __restrict__ a_dst,
                             float* __restrict__ as_, float* __restrict__ ad_,
                             float* __restrict__ mx, float* __restrict__ dn, int C) {
    int idx = blockIdx.x * blockDim.x + threadIdx.x;
    if (idx >= N_NODES * HEADS) return;
    int n = idx >> 2, hd = idx & 3;
    const float* hp  = h + (size_t)n * (HEADS * C) + hd * C;
    const float* avs = a_src + hd * C;
    const float* avd = a_dst + hd * C;
    float s = 0.0f, d = 0.0f;
    for (int c = 0; c < C; ++c) { s = fmaf(hp[c], avs[c], s); d = fmaf(hp[c], avd[c], d); }
    as_[idx] = s; ad_[idx] = d;
    mx[idx] = -__builtin_inff();
    dn[idx] = 0.0f;
}

// ---------------- edge pass 1: leaky-relu logits + segment max ----------------
__global__ void edge_logits_max(const long long* __restrict__ esrc, const long long* __restrict__ edst,
                                const float* __restrict__ as_, const float* __restrict__ ad_,
                                float* __restrict__ ev, float* __restrict__ mx) {
    int e = blockIdx.x * blockDim.x + threadIdx.x;
    if (e >= E_TOT) return;
    long long s = (e < N_EDGES) ? esrc[e] : (long long)(e - N_EDGES);
    long long d = (e < N_EDGES) ? edst[e] : (long long)(e - N_EDGES);
#pragma unroll
    for (int h = 0; h < HEADS; ++h) {
        float z = as_[s * HEADS + h] + ad_[d * HEADS + h];
        z = (z > 0.0f) ? z : NEG_SLOPE * z;
        ev[(size_t)e * HEADS + h] = z;
        atomicMaxFloat(&mx[d * HEADS + h], z);
    }
}

// ---------------- edge pass 2: exp + segment sum ----------------
__global__ void edge_exp(const long long* __restrict__ edst,
                         float* __restrict__ ev, const float* __restrict__ mx,
                         float* __restrict__ dn) {
    int e = blockIdx.x * blockDim.x + threadIdx.x;
    if (e >= E_TOT) return;
    long long d = (e < N_EDGES) ? edst[e] : (long long)(e - N_EDGES);
#pragma unroll
    for (int h = 0; h < HEADS; ++h) {
        float p = expf(ev[(size_t)e * HEADS + h] - mx[d * HEADS + h]);
        ev[(size_t)e * HEADS + h] = p;
        atomicAdd(&dn[d * HEADS + h], p);
    }
}

// ---------------- edge pass 3: weighted message aggregation (wave32 per edge) ----------------
template<int C>   // channels per head; total HC = 4*C; each lane handles HC/32 channels
__global__ void __launch_bounds__(256) edge_aggregate(const long long* __restrict__ esrc,
                                                      const long long* __restrict__ edst,
                                                      const float* __restrict__ hfeat,
                                                      const float* __restrict__ p,
                                                      const float* __restrict__ dn,
                                                      float* __restrict__ out) {
    const int HC  = HEADS * C;
    const int PER = HC / 32;
    int wave = blockIdx.x * (blockDim.x >> 5) + (threadIdx.x >> 5);
    if (wave >= E_TOT) return;
    int lane = threadIdx.x & 31;
    long long s = (wave < N_EDGES) ? esrc[wave] : (long long)(wave - N_EDGES);
    long long d = (wave < N_EDGES) ? edst[wave] : (long long)(wave - N_EDGES);
    float al[HEADS];
#pragma unroll
    for (int h = 0; h < HEADS; ++h)
        al[h] = p[(size_t)wave * HEADS + h] / dn[d * HEADS + h];
    const float* hs = hfeat + (size_t)s * HC;
    float*       od = out   + (size_t)d * HC;
    __builtin_prefetch(hs, 0, 1);
#pragma unroll
    for (int i = 0; i < PER; ++i) {
        int c = lane * PER + i;
        atomicAdd(&od[c], hs[c] * al[c / C]);
    }
}

// ---------------- elementwise: ELU(x + bias) in place ----------------
__global__ void elu_bias_kernel(float* __restrict__ h, const float* __restrict__ b, int dim) {
    int idx = blockIdx.x * blockDim.x + threadIdx.x;
    if (idx >= N_NODES * dim) return;
    float v = h[idx] + b[idx % dim];
    h[idx] = (v > 0.0f) ? v : (expf(v) - 1.0f);
}

__global__ void zero_kernel(float* __restrict__ p, int n) {
    int idx = blockIdx.x * blockDim.x + threadIdx.x;
    if (idx < n) p[idx] = 0.0f;
}

__global__ void bias_init_kernel(float* __restrict__ out, const float* __restrict__ b, int dim, int n) {
    int idx = blockIdx.x * blockDim.x + threadIdx.x;
    if (idx < n) out[idx] = b[idx % dim];
}

// ---------------- launcher ----------------
extern "C" void kernel_launch(void* const* d_in, const int* in_sizes, int n_in,
                              void* d_out, int out_size, void* d_ws, size_t ws_size,
                              hipStream_t stream) {
    (void)in_sizes; (void)n_in; (void)out_size; (void)ws_size;
    const float*     x       = (const float*)d_in[0];
    const long long* eidx    = (const long long*)d_in[1];
    const float*     bn_g    = (const float*)d_in[2];
    const float*     bn_b    = (const float*)d_in[3];
    const float*     bn_m    = (const float*)d_in[4];
    const float*     bn_v    = (const float*)d_in[5];
    const float*     W1      = (const float*)d_in[6];
    const float*     a1_src  = (const float*)d_in[7];
    const float*     a1_dst  = (const float*)d_in[8];
    const float*     b1      = (const float*)d_in[9];
    const float*     W2      = (const float*)d_in[10];
    const float*     a2_src  = (const float*)d_in[11];
    const float*     a2_dst  = (const float*)d_in[12];
    const float*     b2      = (const float*)d_in[13];
    float*           out     = (float*)d_out;

    const long long* esrc = eidx;
    const long long* edst = eidx + N_EDGES;

    // workspace carve-up (floats); reuse buffers across the two layers
    float* ws = (float*)d_ws;
    size_t off = 0;
    float* hbn  = ws + off; off += (size_t)N_NODES * IN_PAD;   // BN output, padded K
    float* h1   = ws + off; off += (size_t)N_NODES * H1DIM;    // GEMM1 output
    float* agg1 = ws + off; off += (size_t)N_NODES * H1DIM;    // layer-1 aggregate / GEMM2 input
    float* h2   = ws + off; off += (size_t)N_NODES * H2DIM;    // GEMM2 output
    float* as_  = ws + off; off += (size_t)N_NODES * HEADS;    // reused both layers
    float* ad_  = ws + off; off += (size_t)N_NODES * HEADS;
    float* mx   = ws + off; off += (size_t)N_NODES * HEADS;
    float* dn   = ws + off; off += (size_t)N_NODES * HEADS;
    float* ev   = ws + off; off += (size_t)E_TOT   * HEADS;    // logits -> p, reused
    float* W1p  = ws + off; off += (size_t)IN_PAD  * H1DIM;    // zero-padded W1 (132x128)

    const int T = 256;
    auto cdiv = [](long long a, long long b) { return (int)((a + b - 1) / b); };

    // ---- layer 1 ----
    bn_pad_kernel<<<cdiv((long long)N_NODES * IN_PAD, T), T, 0, stream>>>(x, bn_g, bn_b, bn_m, bn_v, hbn);
    pad_w_kernel<<<cdiv((long long)IN_PAD * H1DIM, T), T, 0, stream>>>(W1, W1p);
    gemm_wmma<IN_PAD, H1DIM><<<N_NODES / 16, 128, 0, stream>>>(hbn, W1p, h1);
    alpha_kernel<<<cdiv((long long)N_NODES * HEADS, T), T, 0, stream>>>(h1, a1_src, a1_dst, as_, ad_, mx, dn, HID);
    zero_kernel<<<cdiv((long long)N_NODES * H1DIM, T), T, 0, stream>>>(agg1, N_NODES * H1DIM);
    edge_logits_max<<<cdiv(E_TOT, T), T, 0, stream>>>(esrc, edst, as_, ad_, ev, mx);
    edge_exp<<<cdiv(E_TOT, T), T, 0, stream>>>(edst, ev, mx, dn);
    edge_aggregate<HID><<<cdiv(E_TOT, 8), 256, 0, stream>>>(esrc, edst, h1, ev, dn, agg1);
    elu_bias_kernel<<<cdiv((long long)N_NODES * H1DIM, T), T, 0, stream>>>(agg1, b1, H1DIM);

    // ---- layer 2 ---- (W2 is 128 = K rows already, no padding needed)
    gemm_wmma<H1DIM, H2DIM><<<N_NODES / 16, 128, 0, stream>>>(agg1, W2, h2);
    alpha_kernel<<<cdiv((long long)N_NODES * HEADS, T), T, 0, stream>>>(h2, a2_src, a2_dst, as_, ad_, mx, dn, N_CLS);
    bias_init_kernel<<<cdiv((long long)N_NODES * H2DIM, T), T, 0, stream>>>(out, b2, H2DIM, N_NODES * H2DIM);
    edge_logits_max<<<cdiv(E_TOT, T), T, 0, stream>>>(esrc, edst, as_, ad_, ev, mx);
    edge_exp<<<cdiv(E_TOT, T), T, 0, stream>>>(edst, ev, mx, dn);
    edge_aggregate<N_CLS><<<cdiv(E_TOT, 8), 256, 0, stream>>>(esrc, edst, h2, ev, dn, out);
}